// AdaptiveSpectralGNN_34024730919241
// MI455X (gfx1250) — compile-verified
//
#include <hip/hip_runtime.h>
#include <hip/hip_bf16.h>

#define NN 100000
#define EE 1600000
#define HH 128
#define GG 512
#define CC 12
#define LL 4

typedef float v2f __attribute__((ext_vector_type(2)));
typedef float v8f __attribute__((ext_vector_type(8)));
typedef unsigned int v4u __attribute__((ext_vector_type(4)));
typedef int v4i __attribute__((ext_vector_type(4)));
typedef int v8i __attribute__((ext_vector_type(8)));

// ---------------- utility kernels ----------------

__global__ void k_fill(float* __restrict__ p, float v, long long n) {
  long long i = (long long)blockIdx.x * blockDim.x + threadIdx.x;
  long long stride = (long long)gridDim.x * blockDim.x;
  for (; i < n; i += stride) p[i] = v;
}

__global__ void k_degree(const int* __restrict__ dst, float* __restrict__ deg, int ecnt) {
  int e = blockIdx.x * blockDim.x + threadIdx.x;
  if (e < ecnt) unsafeAtomicAdd(&deg[dst[e]], 1.0f);
}

__global__ void k_rsqrt_inplace(float* __restrict__ d, int n) {
  int i = blockIdx.x * blockDim.x + threadIdx.x;
  if (i < n) d[i] = rsqrtf(d[i]);
}

// ---------------- TDM tile load (ISA §8.3/§8.4 descriptor packing) ----------------
// Loads a 2D tile (tile1 rows x tile0 elems, f32) from a row-major tensor with row
// stride `stride0` (elems) into LDS at byte offset lds_addr, adding pad DWORDs of
// LDS padding after every interval DWORDs (encoded codes). Rows/cols beyond
// dim0_rem/dim1_rem (remaining tensor extent from tile start) read as zero.
__device__ __forceinline__ void tdm_load_tile_f32(
    unsigned lds_addr, const float* gptr,
    unsigned dim0_rem, unsigned dim1_rem,
    unsigned tile0, unsigned tile1, unsigned stride0,
    unsigned pad_interval_code, unsigned pad_amount_code) {
  unsigned long long ga = (unsigned long long)(uintptr_t)gptr;
  v4u g0;
  g0.x = 1u;                                   // count=1, user-mode descriptor
  g0.y = lds_addr;                             // LDS byte address
  g0.z = (unsigned)ga;                         // global_addr[31:0]
  g0.w = (unsigned)(ga >> 32) | (2u << 30);    // global_addr[56:32] | type=2
  v8i g1;
  g1[0] = (int)((2u << 16)                     // data_size = 4 bytes
                | (1u << 20)                   // pad_enable
                | (pad_interval_code << 22)
                | (pad_amount_code << 25));
  g1[1] = (int)((dim0_rem & 0xFFFFu) << 16);                     // tensor_dim0[15:0]
  g1[2] = (int)((dim0_rem >> 16) | ((dim1_rem & 0xFFFFu) << 16));// dim0 hi | dim1 lo
  g1[3] = (int)((dim1_rem >> 16) | (tile0 << 16));               // dim1 hi | tile_dim0
  g1[4] = (int)(tile1);                                          // tile_dim1 (tile_dim2=0)
  g1[5] = (int)stride0;                                          // tensor_dim0_stride lo32
  g1[6] = 0;
  g1[7] = 0;
  v4i z4 = {0, 0, 0, 0};
  v8i z8 = {0, 0, 0, 0, 0, 0, 0, 0};
  __builtin_amdgcn_tensor_load_to_lds(g0, g1, z4, z4, z8, 0);
}

// ---------------- WMMA f32 GEMM: out[M x 128] = A[M x 128] @ W[128 x 128] ----------------
// 256 threads = 8 wave32 waves; each wave owns a 16x128 strip (8 tiles of 16x16, f32
// accumulators). K tiled by 32; tiles staged by the Tensor Data Mover directly into
// LDS (double-buffered, TENSORcnt-synchronized), compute via V_WMMA_F32_16X16X4_F32.

#define A_BUF_FLOATS (128 * 33)   // stride 33: conflict-free A-fragment reads
#define W_BUF_FLOATS (32 * 136)   // stride 136: conflict-free B-fragment reads

__global__ __launch_bounds__(256)
void k_gemm128(const float* __restrict__ A, const float* __restrict__ W,
               const float* __restrict__ bias, float* __restrict__ out,
               int M, int do_relu) {
  __shared__ float smem[2 * A_BUF_FLOATS + 2 * W_BUF_FLOATS];

  const int tid = threadIdx.x;
  const int blockRow = blockIdx.x * 128;
  const int lane = tid & 31;
  const int half = lane >> 4;            // lane-half selector (ISA fragment layout)
  const int l16  = lane & 15;
  const int warpRow = (tid >> 5) << 4;   // 0..112

  const unsigned ldsBase = (unsigned)(unsigned long long)(uintptr_t)&smem[0];
  const unsigned ldsA0 = ldsBase;
  const unsigned ldsA1 = ldsBase + A_BUF_FLOATS * 4u;
  const unsigned ldsW0 = ldsBase + 2u * A_BUF_FLOATS * 4u;
  const unsigned ldsW1 = ldsW0 + W_BUF_FLOATS * 4u;
  const unsigned rowsRem = (unsigned)(M - blockRow);   // > 0 by grid sizing

  v8f acc[8];
  #pragma unroll
  for (int nt = 0; nt < 8; ++nt)
    #pragma unroll
    for (int j = 0; j < 8; ++j) acc[nt][j] = 0.0f;

  // prologue: stage k0 = 0 into buffer 0 (wave 0 drives the TDM)
  if (tid < 32) {
    // A tile: 128 rows x 32 cols, row stride 128; LDS rows padded 32+1 DW
    tdm_load_tile_f32(ldsA0, A + (size_t)blockRow * 128, 128u, rowsRem,
                      32u, 128u, 128u, /*interval 32DW*/4u, /*pad 1DW*/0u);
    // W tile: 32 rows x 128 cols, row stride 128; LDS rows padded 128+8 DW
    tdm_load_tile_f32(ldsW0, W, 128u, 128u,
                      128u, 32u, 128u, /*interval 128DW*/6u, /*pad 8DW*/7u);
  }

  for (int s = 0; s < 4; ++s) {
    const int k0 = s * 32;
    const int buf = s & 1;
    if (tid < 32) {
      if (s < 3) {
        const int k1 = k0 + 32;
        tdm_load_tile_f32(buf ? ldsA0 : ldsA1,
                          A + (size_t)blockRow * 128 + k1,
                          (unsigned)(128 - k1), rowsRem,
                          32u, 128u, 128u, 4u, 0u);
        tdm_load_tile_f32(buf ? ldsW0 : ldsW1,
                          W + (size_t)k1 * 128,
                          128u, (unsigned)(128 - k1),
                          128u, 32u, 128u, 6u, 7u);
        __builtin_amdgcn_s_wait_tensorcnt(2);  // current stage landed (in-order)
      } else {
        __builtin_amdgcn_s_wait_tensorcnt(0);
      }
    }
    __syncthreads();  // publish TDM-written LDS to all waves

    const float* lA = smem + (buf ? A_BUF_FLOATS : 0);
    const float* lW = smem + 2 * A_BUF_FLOATS + (buf ? W_BUF_FLOATS : 0);

    #pragma unroll
    for (int kk = 0; kk < 32; kk += 4) {
      // A fragment 16x4: lanes 0-15 hold K=kk,kk+1 ; lanes 16-31 hold K=kk+2,kk+3
      v2f a;
      const int ai = (warpRow + l16) * 33 + kk + 2 * half;
      a.x = lA[ai];
      a.y = lA[ai + 1];
      #pragma unroll
      for (int nt = 0; nt < 8; ++nt) {
        // B fragment 4x16: rows striped across lanes; K split across lane halves
        v2f b;
        const int bi = (kk + 2 * half) * 136 + nt * 16 + l16;
        b.x = lW[bi];
        b.y = lW[bi + 136];
        acc[nt] = __builtin_amdgcn_wmma_f32_16x16x4_f32(
            false, a, false, b, (short)0, acc[nt], false, false);
      }
    }
    __syncthreads();  // done reading this buffer before TDM overwrites it
  }

  // store: C/D layout — VGPR i: lanes0-15 -> M=i, lanes16-31 -> M=i+8
  #pragma unroll
  for (int nt = 0; nt < 8; ++nt) {
    int col = nt * 16 + l16;
    float bv = bias ? bias[col] : 0.0f;
    #pragma unroll
    for (int i = 0; i < 8; ++i) {
      int grow = blockRow + warpRow + i + 8 * half;
      if (grow < M) {
        float v = acc[nt][i] + bv;
        if (do_relu) v = fmaxf(v, 0.0f);
        out[(size_t)grow * 128 + col] = v;
      }
    }
  }
}

// ---------------- GCN aggregation ----------------

// agg[n,:] = hw[n,:] * dinv[n]^2   (self-loop message, initializes agg)
__global__ void k_selfloop(const float* __restrict__ hw, const float* __restrict__ dinv,
                           float* __restrict__ agg, int n_nodes) {
  long long t = (long long)blockIdx.x * blockDim.x + threadIdx.x;
  int n = (int)(t >> 5);
  if (n >= n_nodes) return;
  int lane = (int)(t & 31);
  float di = dinv[n];
  float c = di * di;
  float4 v = *(const float4*)(hw + (size_t)n * 128 + lane * 4);
  v.x *= c; v.y *= c; v.z *= c; v.w *= c;
  *(float4*)(agg + (size_t)n * 128 + lane * 4) = v;
}

// one wave per edge: agg[dst,:] += coef * hw[src,:]
__global__ void k_scatter(const int* __restrict__ src, const int* __restrict__ dst,
                          const float* __restrict__ dinv, const float* __restrict__ hw,
                          float* __restrict__ agg, int ecnt) {
  long long t = (long long)blockIdx.x * blockDim.x + threadIdx.x;
  int e = (int)(t >> 5);
  if (e >= ecnt) return;
  int lane = (int)(t & 31);
  int s = src[e], d = dst[e];
  float coef = dinv[s] * dinv[d];
  float4 v = *(const float4*)(hw + (size_t)s * 128 + lane * 4);
  float* ar = agg + (size_t)d * 128 + lane * 4;
  unsafeAtomicAdd(ar + 0, v.x * coef);
  unsafeAtomicAdd(ar + 1, v.y * coef);
  unsafeAtomicAdd(ar + 2, v.z * coef);
  unsafeAtomicAdd(ar + 3, v.w * coef);
}

// h = relu(BN(agg + conv_b)) in place
__global__ void k_bnrelu(float* __restrict__ h,
                         const float* __restrict__ cb, const float* __restrict__ gamma,
                         const float* __restrict__ beta, const float* __restrict__ mean,
                         const float* __restrict__ var, int n_nodes) {
  long long idx = (long long)blockIdx.x * blockDim.x + threadIdx.x;
  if (idx >= (long long)n_nodes * 128) return;
  int f = (int)(idx & 127);
  float v = h[idx] + cb[f];
  v = (v - mean[f]) * rsqrtf(var[f] + 1e-5f) * gamma[f] + beta[f];
  h[idx] = fmaxf(v, 0.0f);
}

// ---------------- pooling + MLP head ----------------

__global__ void k_pool(const float* __restrict__ h, const int* __restrict__ batch,
                       float* __restrict__ sums, float* __restrict__ cnts, int n_nodes) {
  long long t = (long long)blockIdx.x * blockDim.x + threadIdx.x;
  int n = (int)(t >> 5);
  if (n >= n_nodes) return;
  int lane = (int)(t & 31);
  int b = batch[n];
  float4 v = *(const float4*)(h + (size_t)n * 128 + lane * 4);
  float* sr = sums + (size_t)b * 128 + lane * 4;
  unsafeAtomicAdd(sr + 0, v.x);
  unsafeAtomicAdd(sr + 1, v.y);
  unsafeAtomicAdd(sr + 2, v.z);
  unsafeAtomicAdd(sr + 3, v.w);
  if (lane == 0) unsafeAtomicAdd(&cnts[b], 1.0f);
}

__global__ void k_head(const float* __restrict__ sums, const float* __restrict__ cnts,
                       const float* __restrict__ fc1w, const float* __restrict__ fc1b,
                       const float* __restrict__ fc2w, const float* __restrict__ fc2b,
                       float* __restrict__ out, int out_elems) {
  __shared__ float gvec[128];
  __shared__ float t1[64];
  int g = blockIdx.x;
  int t = threadIdx.x;
  float cnt = fmaxf(cnts[g], 1.0f);
  if (t < 128) gvec[t] = sums[(size_t)g * 128 + t] / cnt;
  __syncthreads();
  if (t < 64) {
    float a = fc1b[t];
    #pragma unroll 4
    for (int k = 0; k < 128; ++k) a += gvec[k] * fc1w[k * 64 + t];
    t1[t] = fmaxf(a, 0.0f);
  }
  __syncthreads();
  if (t < CC) {
    float a = fc2b[t];
    #pragma unroll 4
    for (int k = 0; k < 64; ++k) a += t1[k] * fc2w[k * CC + t];
    out[g * CC + t] = a;
  }
  // tuple's second element (scalar 0.0) concatenated after the logits
  if (g == 0 && t == CC && out_elems > GG * CC) out[GG * CC] = 0.0f;
}

// ---------------- launch ----------------

extern "C" void kernel_launch(void* const* d_in, const int* in_sizes, int n_in,
                              void* d_out, int out_size, void* d_ws, size_t ws_size,
                              hipStream_t stream) {
  (void)in_sizes; (void)n_in; (void)ws_size;
  const float* x      = (const float*)d_in[0];
  const int*   edge   = (const int*)d_in[1];
  const int*   batch  = (const int*)d_in[2];
  const float* w_in   = (const float*)d_in[3];
  const float* b_in   = (const float*)d_in[4];
  const float* conv_w = (const float*)d_in[5];
  const float* conv_b = (const float*)d_in[6];
  const float* bn_g   = (const float*)d_in[7];
  const float* bn_b   = (const float*)d_in[8];
  const float* bn_m   = (const float*)d_in[9];
  const float* bn_v   = (const float*)d_in[10];
  const float* fc1w   = (const float*)d_in[11];
  const float* fc1b   = (const float*)d_in[12];
  const float* fc2w   = (const float*)d_in[13];
  const float* fc2b   = (const float*)d_in[14];
  float* out = (float*)d_out;

  const int* src = edge;
  const int* dst = edge + EE;

  char* ws = (char*)d_ws;
  size_t off = 0;
  auto take = [&](size_t bytes) -> float* {
    float* p = (float*)(ws + off);
    off += (bytes + 255) & ~(size_t)255;
    return p;
  };
  float* dinv = take((size_t)NN * 4);
  float* bufA = take((size_t)NN * HH * 4);
  float* bufB = take((size_t)NN * HH * 4);
  float* bufC = take((size_t)NN * HH * 4);
  float* sums = take((size_t)GG * HH * 4);
  float* cnts = take((size_t)GG * 4);

  // degree (self-loop => start at 1), then dinv = rsqrt(deg)
  k_fill<<<1024, 256, 0, stream>>>(dinv, 1.0f, (long long)NN);
  k_degree<<<(EE + 255) / 256, 256, 0, stream>>>(dst, dinv, EE);
  k_rsqrt_inplace<<<(NN + 255) / 256, 256, 0, stream>>>(dinv, NN);

  const int gblocks = (NN + 127) / 128;
  // input projection + ReLU
  k_gemm128<<<gblocks, 256, 0, stream>>>(x, w_in, b_in, bufA, NN, 1);

  float* hcur = bufA;
  float* hnext = bufC;
  for (int l = 0; l < LL; ++l) {
    // hw = h @ W_l
    k_gemm128<<<gblocks, 256, 0, stream>>>(hcur, conv_w + (size_t)l * HH * HH,
                                           nullptr, bufB, NN, 0);
    // agg = hw * dinv^2 (self-loop), then scatter edges, then BN+ReLU in place
    k_selfloop<<<(NN * 32 + 255) / 256, 256, 0, stream>>>(bufB, dinv, hnext, NN);
    k_scatter<<<(EE * 32 + 255) / 256, 256, 0, stream>>>(src, dst, dinv, bufB, hnext, EE);
    k_bnrelu<<<((long long)NN * HH + 255) / 256, 256, 0, stream>>>(
        hnext, conv_b + l * HH, bn_g + l * HH, bn_b + l * HH,
        bn_m + l * HH, bn_v + l * HH, NN);
    float* tmp = hcur; hcur = hnext; hnext = tmp;
  }

  // global mean pool + MLP head
  k_fill<<<256, 256, 0, stream>>>(sums, 0.0f, (long long)GG * HH);
  k_fill<<<8, 256, 0, stream>>>(cnts, 0.0f, (long long)GG);
  k_pool<<<(NN * 32 + 255) / 256, 256, 0, stream>>>(hcur, batch, sums, cnts, NN);
  k_head<<<GG, 128, 0, stream>>>(sums, cnts, fc1w, fc1b, fc2w, fc2b, out, out_size);
}